// HardOptimalCompletionDistillationLoss_3556232921403
// MI455X (gfx1250) — compile-verified
//
#include <hip/hip_runtime.h>
#include <cstdint>

// Problem shape (from reference setup_inputs): T=100, B=64, R=100, C=10000
#define T_DIM 100
#define B_DIM 64
#define R_DIM 100
#define C_DIM 10000

#define BIGI (1 << 28)
#define LOG2E 1.4426950408889634f
#define LN2   0.6931471805599453f

typedef float v2f __attribute__((ext_vector_type(2)));
typedef float v8f __attribute__((ext_vector_type(8)));

// ---------------------------------------------------------------------------
// Kernel A: edit-distance DP (one wave32 per batch) + deterministic dedupe of
// optimal-completion classes. Writes keptCls[(t*B+b)*R + j] = class or -1,
// and cntBuf[t*B+b] = number of distinct optimal classes.
// Row recurrence (costs all 1):
//   a[0] = prev[0]+1 ; a[j] = min(prev[j]+1, prev[j-1] + (ref[j-1]!=h))
//   row[j] = min_{k<=j}(a[k]-k) + j      (prefix-min handles deletions)
// Integer arithmetic is exact (all costs are 1.0).
// ---------------------------------------------------------------------------
__global__ __launch_bounds__(32) void hocd_dp_kernel(const int* __restrict__ ref,
                                                     const int* __restrict__ hyp,
                                                     int* __restrict__ keptCls,
                                                     int* __restrict__ cntBuf) {
  const int b = blockIdx.x;
  const int lane = threadIdx.x;

  __shared__ int sCls[R_DIM];                    // ref classes for this batch
  __shared__ int sCandJ[R_DIM + 4];              // compacted candidate j's
  __shared__ __align__(16) int sKept[R_DIM];     // kept class per j, -1 if none

  for (int j = lane; j < R_DIM; j += 32) sCls[j] = ref[j * B_DIM + b];
  __syncthreads();

  // Each lane owns j = lane*4 + k, k=0..3 (covers j=0..127; valid j<=100).
  int jidx[4], refm1[4], prev[4];
#pragma unroll
  for (int k = 0; k < 4; ++k) {
    const int j = lane * 4 + k;
    jidx[k] = j;
    refm1[k] = (j >= 1 && j <= R_DIM) ? sCls[j - 1] : 0;
    prev[k] = (j <= R_DIM) ? j : BIGI;           // row0[j] = j
  }

  for (int t = 0; t < T_DIM; ++t) {
    if (t > 0) {
      const int h = hyp[(t - 1) * B_DIM + b];
      int pm[4];
      pm[0] = __shfl_up(prev[3], 1);             // prev[j-1] crossing lanes
      pm[1] = prev[0]; pm[2] = prev[1]; pm[3] = prev[2];
      int tv[4];
#pragma unroll
      for (int k = 0; k < 4; ++k) {
        const int j = jidx[k];
        int av;
        if (j == 0)           av = prev[0] + 1;
        else                  av = min(prev[k] + 1, pm[k] + (refm1[k] != h ? 1 : 0));
        if (j > R_DIM)        av = BIGI;         // sentinel lanes never pollute
        tv[k] = av - j;                          // a[j] - jdel[j]
      }
      // inclusive prefix-min over flattened j order: serial in-lane, then wave scan
      const int p0 = tv[0];
      const int p1 = min(p0, tv[1]);
      const int p2 = min(p1, tv[2]);
      const int p3 = min(p2, tv[3]);
      int s = p3;
#pragma unroll
      for (int off = 1; off < 32; off <<= 1) {
        const int o = __shfl_up(s, off);
        if (lane >= off) s = min(s, o);
      }
      int excl = __shfl_up(s, 1);
      if (lane == 0) excl = BIGI;
      prev[0] = min(excl, p0) + jidx[0];
      prev[1] = min(excl, p1) + jidx[1];
      prev[2] = min(excl, p2) + jidx[2];
      prev[3] = min(excl, p3) + jidx[3];
    }

    // dmin over j in [0, R)  (note: j=R excluded, matching d[:, :R, :])
    int lm = BIGI;
#pragma unroll
    for (int k = 0; k < 4; ++k) if (jidx[k] < R_DIM) lm = min(lm, prev[k]);
#pragma unroll
    for (int off = 16; off; off >>= 1) lm = min(lm, __shfl_xor(lm, off));

    // compact candidates (d == dmin) in increasing-j order via exclusive scan
    int cflag[4]; int cl = 0;
#pragma unroll
    for (int k = 0; k < 4; ++k) {
      cflag[k] = (jidx[k] < R_DIM && prev[k] == lm) ? 1 : 0;
      cl += cflag[k];
    }
    int run = cl;
#pragma unroll
    for (int off = 1; off < 32; off <<= 1) {
      const int o = __shfl_up(run, off);
      if (lane >= off) run += o;
    }
    const int base = run - cl;
    const int K = __shfl(run, 31);

    for (int j = lane; j < R_DIM; j += 32) sKept[j] = -1;
    __syncthreads();
    {
      int idx = base;
#pragma unroll
      for (int k = 0; k < 4; ++k) if (cflag[k]) sCandJ[idx++] = jidx[k];
    }
    __syncthreads();

    // dedupe: candidate i kept iff no earlier candidate has the same class
    int myKept = 0;
    for (int i = lane; i < K; i += 32) {
      const int j = sCandJ[i];
      const int c = sCls[j];
      bool dup = false;
      for (int i2 = 0; i2 < i; ++i2)
        if (sCls[sCandJ[i2]] == c) { dup = true; break; }
      if (!dup) { sKept[j] = c; myKept++; }
    }
#pragma unroll
    for (int off = 16; off; off >>= 1) myKept += __shfl_xor(myKept, off);
    __syncthreads();

    const long rowBase = (long)(t * B_DIM + b) * R_DIM;
    if (lane < R_DIM / 4)  // 25 lanes * int4 = 100 ints
      *(int4*)(keptCls + rowBase + lane * 4) = *(const int4*)(sKept + lane * 4);
    if (lane == 0) cntBuf[t * B_DIM + b] = myKept;
    __syncthreads();
  }
}

// ---------------------------------------------------------------------------
// Kernel B: per-(t,b) row logsumexp + selected-class sum + loss.
// 16 rows per block, 256 threads (8 waves).
//   Pass 1: streaming row max (coalesced b128 loads, warms L2).
//   Pass 2: double-buffered ASYNC global->LDS staging of 16x128 logit tiles;
//           raw v_exp_f32 (1 FMA + 1 TRANS per element) in VALU, accumulation
//           on the matrix pipe via f32 WMMA with B = ones
//           (16 row-sums x K=4 per v_wmma_f32_16x16x4_f32).
// ---------------------------------------------------------------------------
#define RPB 16
#define TILE_COLS 128
#define LDS_STRIDE 132                 // +4 float pad: conflict-free A reads
#define NFULL_TILES 78                 // 78*128 = 9984
#define TAIL_START 9984
#define TAIL_COLS 16

__global__ __launch_bounds__(256) void hocd_lse_kernel(const float* __restrict__ logits,
                                                       const int* __restrict__ keptCls,
                                                       const int* __restrict__ cntBuf,
                                                       float* __restrict__ lossBuf) {
  const int tid = threadIdx.x;
  const int lane = tid & 31;
  const int wave = tid >> 5;
  const int r0 = blockIdx.x * RPB;

  __shared__ __align__(16) float sBuf[2][RPB * LDS_STRIDE];
  __shared__ float sRowMax[RPB];
  __shared__ float sTail[RPB];
  __shared__ float sPartial[8 * RPB];
  __shared__ float sSelPart[RPB * 16];

  // ---- Pass 1: row maxima (wave w owns rows 2w, 2w+1) ----
  for (int rr = 0; rr < 2; ++rr) {
    const int row = wave * 2 + rr;
    const float4* rp = (const float4*)(logits + (size_t)(r0 + row) * C_DIM);
    float m = -3.4e38f;
    for (int i = lane; i < C_DIM / 4; i += 32) {
      const float4 v = rp[i];
      m = fmaxf(fmaxf(m, v.x), fmaxf(v.y, fmaxf(v.z, v.w)));
    }
#pragma unroll
    for (int off = 16; off; off >>= 1) m = fmaxf(m, __shfl_xor(m, off));
    if (lane == 0) sRowMax[row] = m;
  }
  __syncthreads();

  // ---- Tail (cols 9984..9999) without WMMA, fixed-order reduction ----
  // exp(x - m) = exp2(x*log2e - m*log2e): 1 fma + 1 raw v_exp_f32. Arguments
  // are <= 0; terms below 2^-126 are irrelevant to a sum >= 1, so the OCML
  // denormal fixup path is unnecessary.
  for (int rr = 0; rr < 2; ++rr) {
    const int row = wave * 2 + rr;
    const float ms = sRowMax[row] * LOG2E;
    float e = 0.f;
    if (lane < TAIL_COLS) {
      const float x = logits[(size_t)(r0 + row) * C_DIM + TAIL_START + lane];
      e = __builtin_amdgcn_exp2f(__builtin_fmaf(x, LOG2E, -ms));
    }
#pragma unroll
    for (int off = 16; off; off >>= 1) e += __shfl_xor(e, off);
    if (lane == 0) sTail[row] = e;
  }

  // ---- Pass 2: async-staged tiles + WMMA f32 accumulation ----
  const int arow = lane & 15;                 // A-matrix M index for this lane
  const float amaxs = sRowMax[arow] * LOG2E;  // pre-scaled row max
  const int koff = (lane >> 4) << 1;          // K offset: 0 (lanes 0-15) / 2 (16-31)
  v8f cacc = {0.f, 0.f, 0.f, 0.f, 0.f, 0.f, 0.f, 0.f};
  const v2f bones = {1.f, 1.f};               // B = ones -> D[m,*] = sum_k A[m,k]

  auto issue_tile = [&](int tile, int bufsel) {
    const int c0 = tile * TILE_COLS;
#pragma unroll
    for (int kk = 0; kk < 2; ++kk) {          // 512 x 16B units, 256 threads
      const int u = tid + kk * 256;
      const int row = u >> 5;                 // 0..15
      const int off16 = u & 31;               // 16B unit within 512B row chunk
      const float* gsrc = logits + (size_t)(r0 + row) * C_DIM + c0 + off16 * 4;
      const float* ldst = &sBuf[bufsel][row * LDS_STRIDE + off16 * 4];
      const uint32_t ldsoff = (uint32_t)(uintptr_t)ldst;
      asm volatile("global_load_async_to_lds_b128 %0, %1, off"
                   :: "v"(ldsoff), "v"(gsrc) : "memory");
    }
  };

  issue_tile(0, 0);
  for (int ti = 0; ti < NFULL_TILES; ++ti) {
    if (ti + 1 < NFULL_TILES) {
      issue_tile(ti + 1, (ti + 1) & 1);
      asm volatile("s_wait_asynccnt 0x2" ::: "memory");   // tile ti landed
    } else {
      asm volatile("s_wait_asynccnt 0x0" ::: "memory");
    }
    __syncthreads();
    const float* buf = sBuf[ti & 1];
#pragma unroll
    for (int kk = 0; kk < 4; ++kk) {          // wave w consumes cols w*16..w*16+15
      const int col = wave * 16 + kk * 4 + koff;
      const float2 xv = *(const float2*)&buf[arow * LDS_STRIDE + col];
      v2f a;
      a.x = __builtin_amdgcn_exp2f(__builtin_fmaf(xv.x, LOG2E, -amaxs));
      a.y = __builtin_amdgcn_exp2f(__builtin_fmaf(xv.y, LOG2E, -amaxs));
      cacc = __builtin_amdgcn_wmma_f32_16x16x4_f32(false, a, false, bones,
                                                   (short)0, cacc, false, false);
    }
    __syncthreads();
  }

  // D layout: lane 0 holds rows 0..7 (N=0), lane 16 holds rows 8..15 (N=0)
  if (lane == 0) {
#pragma unroll
    for (int v = 0; v < 8; ++v) sPartial[wave * 16 + v] = cacc[v];
  }
  if (lane == 16) {
#pragma unroll
    for (int v = 0; v < 8; ++v) sPartial[wave * 16 + 8 + v] = cacc[v];
  }
  __syncthreads();

  // ---- Gather selected classes (L2-hot after the streams) ----
  {
    const int grow = tid >> 4;                // row 0..15
    const int slot = tid & 15;
    const size_t r = (size_t)(r0 + grow);
    float acc = 0.f;
    for (int j = slot; j < R_DIM; j += 16) {
      const int c = keptCls[r * R_DIM + j];
      if (c >= 0) acc += logits[r * C_DIM + c];
    }
    sSelPart[grow * 16 + slot] = acc;
  }
  __syncthreads();

  // ---- Finalize (fixed-order sums => deterministic) ----
  if (tid < RPB) {
    const int r = r0 + tid;
    float se = sTail[tid];
#pragma unroll
    for (int w = 0; w < 8; ++w) se += sPartial[w * 16 + tid];
    // ln(se) = log2(se) * ln2 ; se >= 1 so raw v_log_f32 is exact enough
    const float lse = sRowMax[tid] + __builtin_amdgcn_logf(se) * LN2;
    float sel = 0.f;
    for (int i = 0; i < 16; ++i) sel += sSelPart[tid * 16 + i];
    const int c = cntBuf[r];
    lossBuf[r] = (c > 0) ? (lse - sel / (float)c) : 0.f;
  }
}

// ---------------------------------------------------------------------------
// Kernel C: deterministic mean of the 6400 per-(t,b) losses.
// ---------------------------------------------------------------------------
__global__ __launch_bounds__(256) void hocd_reduce_kernel(const float* __restrict__ lossBuf,
                                                          float* __restrict__ out) {
  __shared__ float red[256];
  const int tid = threadIdx.x;
  float a = 0.f;
  for (int i = tid; i < T_DIM * B_DIM; i += 256) a += lossBuf[i];
  red[tid] = a;
  __syncthreads();
  for (int s = 128; s > 0; s >>= 1) {
    if (tid < s) red[tid] += red[tid + s];
    __syncthreads();
  }
  if (tid == 0) out[0] = red[0] / (float)(T_DIM * B_DIM);
}

// ---------------------------------------------------------------------------
extern "C" void kernel_launch(void* const* d_in, const int* in_sizes, int n_in,
                              void* d_out, int out_size, void* d_ws, size_t ws_size,
                              hipStream_t stream) {
  (void)in_sizes; (void)n_in; (void)out_size; (void)ws_size;
  const float* logits = (const float*)d_in[0];   // (T,B,C) f32
  const int* ref = (const int*)d_in[1];          // (R,B)   i32
  const int* hyp = (const int*)d_in[2];          // (T,B)   i32
  float* out = (float*)d_out;                    // scalar f32

  // Workspace layout (~2.61 MB): kept classes, counts, per-row losses.
  int* keptCls = (int*)d_ws;                             // T*B*R ints
  int* cntBuf = keptCls + (size_t)T_DIM * B_DIM * R_DIM; // T*B ints
  float* lossBuf = (float*)(cntBuf + T_DIM * B_DIM);     // T*B floats

  hocd_dp_kernel<<<B_DIM, 32, 0, stream>>>(ref, hyp, keptCls, cntBuf);
  hocd_lse_kernel<<<(T_DIM * B_DIM) / RPB, 256, 0, stream>>>(logits, keptCls, cntBuf, lossBuf);
  hocd_reduce_kernel<<<1, 256, 0, stream>>>(lossBuf, out);
}